// SegGCN_14688788152861
// MI455X (gfx1250) — compile-verified
//
#include <hip/hip_runtime.h>
#include <hip/hip_bf16.h>
#include <math.h>

// ---------------- problem constants ----------------
#define BS    8
#define C     2
#define Hh    256
#define Wd    256
#define HW    (Hh * Wd)          // 65536
#define Dd    64
#define NSEG  2048
#define NMASK 1024
#define NAUG  1536
#define NE1   32768
#define NE2   24576
#define NQ    65536              // queue columns
#define F     128                // feature dim after W
#define OUTSTRIDE (8 + NQ)       // 65544 logits row stride
#define LOGITS_ELEMS ((size_t)NMASK * OUTSTRIDE)
#define INV_T (1.0f / 0.07f)

typedef __attribute__((ext_vector_type(2))) float v2f;
typedef __attribute__((ext_vector_type(8))) float v8f;

// ---------------- reductions ----------------
__global__ void bce_kernel(const float* __restrict__ seg, const float* __restrict__ cam,
                           float* __restrict__ acc, int n) {
  __shared__ float sh0[256], sh1[256], sh2[256];
  int tid = threadIdx.x;
  float p = 0.f, lp = 0.f, ln = 0.f;
  for (int i = blockIdx.x * blockDim.x + tid; i < n; i += gridDim.x * blockDim.x) {
    float x = seg[i];
    float lab = (cam[i] >= 0.5f) ? 1.f : 0.f;
    float gtz = (x >= 0.f) ? 1.f : 0.f;
    float lv = x * (lab - gtz) - log1pf(expf(-fabsf(x)));
    p += lab; lp += lab * lv; ln += (1.f - lab) * lv;
  }
  sh0[tid] = p; sh1[tid] = lp; sh2[tid] = ln;
  __syncthreads();
  for (int s = 128; s > 0; s >>= 1) {
    if (tid < s) { sh0[tid] += sh0[tid + s]; sh1[tid] += sh1[tid + s]; sh2[tid] += sh2[tid + s]; }
    __syncthreads();
  }
  if (tid == 0) { atomicAdd(&acc[0], sh0[0]); atomicAdd(&acc[1], sh1[0]); atomicAdd(&acc[2], sh2[0]); }
}

__global__ void loss_final_kernel(const float* __restrict__ acc, float* __restrict__ out) {
  if (threadIdx.x || blockIdx.x) return;
  float total = (float)(BS * C * HW);
  float np = acc[0];
  float nn = total - np;
  float w = nn / (np + nn);
  float loss = (w * (-acc[1]) + (1.0f - w) * (-acc[2])) / total;
  out[LOGITS_ELEMS + NMASK] = loss;
}

__global__ void spmax_kernel(const int* __restrict__ sp, int* __restrict__ smax, int n) {
  __shared__ int sh[256];
  int tid = threadIdx.x;
  int m = 0;
  for (int i = blockIdx.x * blockDim.x + tid; i < n; i += gridDim.x * blockDim.x)
    m = max(m, sp[i]);
  sh[tid] = m; __syncthreads();
  for (int s = 128; s > 0; s >>= 1) { if (tid < s) sh[tid] = max(sh[tid], sh[tid + s]); __syncthreads(); }
  if (tid == 0) atomicMax(smax, sh[0]);
}

// ---------------- unique / searchsorted emulation ----------------
__global__ void presence_kernel(const int* __restrict__ sp, const int* __restrict__ smax,
                                int* __restrict__ pres) {
  int i = blockIdx.x * blockDim.x + threadIdx.x;
  if (i >= BS * HW) return;
  int M = smax[0] + 1;
  int b = i / HW;
  int v = sp[i] + M * b;
  if (v >= 0 && v < NSEG) pres[v] = 1;
}

__global__ void rank_scan_kernel(const int* __restrict__ pres, int* __restrict__ rank) {
  if (threadIdx.x || blockIdx.x) return;
  int r = 0;
  for (int v = 0; v < NSEG; ++v) { rank[v] = r; r += pres[v]; }
}

// ---------------- scatter means ----------------
__global__ void emb_scatter_kernel(const float* __restrict__ emb, const int* __restrict__ sp,
                                   const int* __restrict__ smax, const int* __restrict__ rank,
                                   float* __restrict__ seg_sum) {
  int t = blockIdx.x * blockDim.x + threadIdx.x;   // [0, BS*Dd*HW)
  if (t >= BS * Dd * HW) return;
  int b = t / (Dd * HW);
  int rem = t - b * (Dd * HW);
  int d = rem / HW;
  int p = rem - d * HW;
  int M = smax[0] + 1;
  int s = rank[sp[b * HW + p] + M * b];
  atomicAdd(&seg_sum[s * Dd + d], emb[t]);
}

__global__ void cnt_scatter_kernel(const int* __restrict__ sp, const int* __restrict__ smax,
                                   const int* __restrict__ rank, const float* __restrict__ cam,
                                   float* __restrict__ seg_cnt, float* __restrict__ cam1_sum) {
  int i = blockIdx.x * blockDim.x + threadIdx.x;
  if (i >= BS * HW) return;
  int b = i / HW;
  int p = i - b * HW;
  int M = smax[0] + 1;
  int s = rank[sp[i] + M * b];
  atomicAdd(&seg_cnt[s], 1.0f);
  atomicAdd(&cam1_sum[s], cam[(b * C + 1) * HW + p]);
}

__global__ void mean_kernel(const float* __restrict__ sum, const float* __restrict__ cnt,
                            float* __restrict__ out, int rows, int cols) {
  int i = blockIdx.x * blockDim.x + threadIdx.x;
  if (i >= rows * cols) return;
  out[i] = sum[i] / fmaxf(cnt[i / cols], 1.0f);
}

// ---------------- GCN pieces ----------------
__global__ void edge_scatter_kernel(const int* __restrict__ edge, int nedges,
                                    const float* __restrict__ x, float* __restrict__ agg_sum,
                                    float* __restrict__ agg_cnt) {
  int t = blockIdx.x * blockDim.x + threadIdx.x;
  if (t >= nedges * Dd) return;
  int e = t / Dd, d = t - e * Dd;
  int src = edge[e], dst = edge[nedges + e];
  atomicAdd(&agg_sum[dst * Dd + d], x[src * Dd + d]);
  if (d == 0) atomicAdd(&agg_cnt[dst], 1.0f);
}

// one block (128 threads) per row: xa = x + agg_mean ; y = relu(xa @ W + b); optional L2-normalize
__global__ __launch_bounds__(128) void gcn_row_kernel(const float* __restrict__ x,
    const float* __restrict__ agg_sum, const float* __restrict__ agg_cnt,
    const float* __restrict__ Wm, const float* __restrict__ bias,
    float* __restrict__ out, int normalize) {
  __shared__ float xr[Dd];
  __shared__ float red[F];
  int row = blockIdx.x;
  int j = threadIdx.x;
  if (j < Dd) {
    float c = fmaxf(agg_cnt[row], 1.0f);
    xr[j] = x[row * Dd + j] + agg_sum[row * Dd + j] / c;
  }
  __syncthreads();
  float acc = bias[j];
  #pragma unroll 8
  for (int k = 0; k < Dd; ++k) acc = fmaf(xr[k], Wm[k * F + j], acc);
  acc = fmaxf(acc, 0.0f);
  if (normalize) {
    red[j] = acc * acc;
    __syncthreads();
    for (int s = 64; s > 0; s >>= 1) { if (j < s) red[j] += red[j + s]; __syncthreads(); }
    acc *= rsqrtf(red[0]);
  }
  out[row * F + j] = acc;
}

// ---------------- masks / gathers ----------------
__global__ void camoff_kernel(const float* __restrict__ cam1_sum, const float* __restrict__ seg_cnt,
                              const int* __restrict__ batch_info, int* __restrict__ off) {
  int s = blockIdx.x * blockDim.x + threadIdx.x;
  if (s >= NSEG) return;
  float mean = cam1_sum[s] / fmaxf(seg_cnt[s], 1.0f);
  int csp = (mean > 0.5f) ? 1 : 0;
  off[s] = (csp + batch_info[s]) * csp;
}

__global__ void mask_scan_kernel(const int* __restrict__ off, int* __restrict__ mask) {
  if (threadIdx.x || blockIdx.x) return;
  int c = 0;
  for (int s = 0; s < NSEG && c < NMASK; ++s)
    if (off[s] != 0) mask[c++] = s;
  for (; c < NMASK; ++c) mask[c] = 0;
}

__global__ void xaug_gather_kernel(const float* __restrict__ feats, const long long* __restrict__ keep,
                                   float* __restrict__ xaug) {
  int t = blockIdx.x * blockDim.x + threadIdx.x;
  if (t >= NAUG * Dd) return;
  int r = t / Dd, d = t - r * Dd;
  xaug[t] = feats[(int)keep[r] * Dd + d];
}

__global__ void pool_scatter_kernel(const float* __restrict__ haug, const long long* __restrict__ keep,
                                    const int* __restrict__ off, float* __restrict__ pool_sum,
                                    float* __restrict__ pool_cnt) {
  int t = blockIdx.x * blockDim.x + threadIdx.x;
  if (t >= NAUG * F) return;
  int r = t / F, d = t - r * F;
  int s = off[(int)keep[r]];   // 0..BS
  atomicAdd(&pool_sum[s * F + d], haug[t]);
  if (d == 0) atomicAdd(&pool_cnt[s], 1.0f);
}

__global__ __launch_bounds__(128) void feataug_kernel(const float* __restrict__ pool_sum,
    const float* __restrict__ pool_cnt, float* __restrict__ feat_aug) {
  __shared__ float red[F];
  int i = blockIdx.x;   // 0..BS-1
  int j = threadIdx.x;
  float v = pool_sum[(i + 1) * F + j] / fmaxf(pool_cnt[i + 1], 1.0f);
  red[j] = v * v;
  __syncthreads();
  for (int s = 64; s > 0; s >>= 1) { if (j < s) red[j] += red[j + s]; __syncthreads(); }
  feat_aug[i * F + j] = v * rsqrtf(red[0]);
}

__global__ void qgather_kernel(const float* __restrict__ feat_ori, const int* __restrict__ mask,
                               float* __restrict__ q) {
  int t = blockIdx.x * blockDim.x + threadIdx.x;
  if (t >= NMASK * F) return;
  q[t] = feat_ori[mask[t / F] * F + (t & (F - 1))];
}

__global__ void small_logits_kernel(const float* __restrict__ q, const float* __restrict__ feat_aug,
                                    float* __restrict__ out) {
  int t = blockIdx.x * blockDim.x + threadIdx.x;
  if (t >= NMASK * BS) return;
  int m = t / BS, j = t - m * BS;
  float acc = 0.f;
  #pragma unroll 8
  for (int k = 0; k < F; ++k) acc = fmaf(q[m * F + k], feat_aug[j * F + k], acc);
  out[(size_t)m * OUTSTRIDE + j] = acc * INV_T;
}

__global__ void camred_kernel(const int* __restrict__ off, const int* __restrict__ mask,
                              float* __restrict__ out) {
  int m = blockIdx.x * blockDim.x + threadIdx.x;
  if (m >= NMASK) return;
  out[LOGITS_ELEMS + m] = (float)(off[mask[m]] - 1);
}

// ---------------- big logits GEMM: fp32 WMMA ----------------
// out[m][8+n] = invT * sum_k q[m][k] * queue[k][n]   (m<1024, n<65536, k<128)
// One wave computes a 16(M) x 64(N) tile: 4 N-subtiles sharing the A fragment,
// K loop of 128 in steps of 4 -> 32 x 4 = 128 V_WMMA_F32_16X16X4_F32 per wave.
// A 16x4 f32 layout (ISA 7.12.2): lanes 0-15 hold (K=k0,k0+1), lanes 16-31 (K=k0+2,k0+3),
// row = m0 + (lane&15).  B 4x16 assumed symmetric (row striped across lanes per VGPR).
__global__ __launch_bounds__(256) void logits_wmma_kernel(const float* __restrict__ q,
    const float* __restrict__ queue, float* __restrict__ out) {
  const int lane = threadIdx.x & 31;
  const int wave = threadIdx.x >> 5;
  const int half = lane >> 4;        // 0 | 1
  const int l16  = lane & 15;
  const int m0 = blockIdx.x * 16;                      // 64 M tiles
  const int n0 = (blockIdx.y * 8 + wave) * 64;         // 1024 wave columns * 64

  v8f c0 = {}, c1 = {}, c2 = {}, c3 = {};
  const float* qrow = q + (m0 + l16) * F;

  #pragma unroll 4
  for (int k0 = 0; k0 < F; k0 += 4) {
    v2f a;
    {
      const float2 av = *(const float2*)(qrow + k0 + half * 2);
      a.x = av.x; a.y = av.y;
    }
    const float* bq = queue + (size_t)(k0 + half * 2) * NQ + n0 + l16;
    v2f b0, b1, b2, b3;
    b0.x = bq[0];  b0.y = bq[NQ];
    b1.x = bq[16]; b1.y = bq[16 + NQ];
    b2.x = bq[32]; b2.y = bq[32 + NQ];
    b3.x = bq[48]; b3.y = bq[48 + NQ];
    c0 = __builtin_amdgcn_wmma_f32_16x16x4_f32(false, a, false, b0, (short)0, c0, false, false);
    c1 = __builtin_amdgcn_wmma_f32_16x16x4_f32(false, a, false, b1, (short)0, c1, false, false);
    c2 = __builtin_amdgcn_wmma_f32_16x16x4_f32(false, a, false, b2, (short)0, c2, false, false);
    c3 = __builtin_amdgcn_wmma_f32_16x16x4_f32(false, a, false, b3, (short)0, c3, false, false);
  }

  #pragma unroll
  for (int r = 0; r < 8; ++r) {
    size_t base = (size_t)(m0 + r + half * 8) * OUTSTRIDE + 8 + n0 + l16;
    out[base +  0] = c0[r] * INV_T;
    out[base + 16] = c1[r] * INV_T;
    out[base + 32] = c2[r] * INV_T;
    out[base + 48] = c3[r] * INV_T;
  }
}

// ---------------- launcher ----------------
extern "C" void kernel_launch(void* const* d_in, const int* in_sizes, int n_in,
                              void* d_out, int out_size, void* d_ws, size_t ws_size,
                              hipStream_t stream) {
  const float*     seg   = (const float*)d_in[0];
  const float*     emb   = (const float*)d_in[1];
  const float*     cam   = (const float*)d_in[2];
  const int*       spseg = (const int*)d_in[3];
  const int*       e1    = (const int*)d_in[4];
  const long long* keep  = (const long long*)d_in[5];   // int64
  const int*       e2    = (const int*)d_in[6];
  const int*       binfo = (const int*)d_in[7];
  const float*     queue = (const float*)d_in[8];
  const float*     Wm    = (const float*)d_in[9];
  const float*     bias  = (const float*)d_in[10];
  float* out = (float*)d_out;

  char* ws = (char*)d_ws;
  size_t o = 0;
  auto alloc = [&](size_t bytes) { size_t cur = o; o += (bytes + 255) & ~(size_t)255; return cur; };

  float* acc       = (float*)(ws + alloc(4 * sizeof(float)));
  int*   smax      = (int*)  (ws + alloc(sizeof(int)));
  int*   pres      = (int*)  (ws + alloc(NSEG * sizeof(int)));
  int*   rank      = (int*)  (ws + alloc(NSEG * sizeof(int)));
  float* seg_sum   = (float*)(ws + alloc((size_t)NSEG * Dd * sizeof(float)));
  float* seg_cnt   = (float*)(ws + alloc(NSEG * sizeof(float)));
  float* cam1_sum  = (float*)(ws + alloc(NSEG * sizeof(float)));
  float* feats     = (float*)(ws + alloc((size_t)NSEG * Dd * sizeof(float)));
  float* agg_sum   = (float*)(ws + alloc((size_t)NSEG * Dd * sizeof(float)));
  float* agg_cnt   = (float*)(ws + alloc(NSEG * sizeof(float)));
  float* feat_ori  = (float*)(ws + alloc((size_t)NSEG * F * sizeof(float)));
  int*   offv      = (int*)  (ws + alloc(NSEG * sizeof(int)));
  int*   mask      = (int*)  (ws + alloc(NMASK * sizeof(int)));
  float* xaug      = (float*)(ws + alloc((size_t)NAUG * Dd * sizeof(float)));
  float* agg2_sum  = (float*)(ws + alloc((size_t)NAUG * Dd * sizeof(float)));
  float* agg2_cnt  = (float*)(ws + alloc(NAUG * sizeof(float)));
  float* haug      = (float*)(ws + alloc((size_t)NAUG * F * sizeof(float)));
  float* pool_sum  = (float*)(ws + alloc((size_t)(BS + 1) * F * sizeof(float)));
  float* pool_cnt  = (float*)(ws + alloc((BS + 1) * sizeof(float)));
  float* feat_aug  = (float*)(ws + alloc((size_t)BS * F * sizeof(float)));
  float* qbuf      = (float*)(ws + alloc((size_t)NMASK * F * sizeof(float)));

  size_t zero_bytes = o < ws_size ? o : ws_size;
  hipMemsetAsync(d_ws, 0, zero_bytes, stream);

  const int TB = 256;
  // 1) BCE reduction
  bce_kernel<<<2048, TB, 0, stream>>>(seg, cam, acc, BS * C * HW);
  // 2) sp_seg max
  spmax_kernel<<<1024, TB, 0, stream>>>(spseg, smax, BS * HW);
  // 3) unique/searchsorted emulation
  presence_kernel<<<(BS * HW + TB - 1) / TB, TB, 0, stream>>>(spseg, smax, pres);
  rank_scan_kernel<<<1, 32, 0, stream>>>(pres, rank);
  // 4) segment feature scatter-mean
  emb_scatter_kernel<<<(BS * Dd * HW) / TB, TB, 0, stream>>>(emb, spseg, smax, rank, seg_sum);
  cnt_scatter_kernel<<<(BS * HW) / TB, TB, 0, stream>>>(spseg, smax, rank, cam, seg_cnt, cam1_sum);
  mean_kernel<<<(NSEG * Dd) / TB, TB, 0, stream>>>(seg_sum, seg_cnt, feats, NSEG, Dd);
  // 5) GCN #1 -> feat_ori (normalized)
  edge_scatter_kernel<<<(NE1 * Dd) / TB, TB, 0, stream>>>(e1, NE1, feats, agg_sum, agg_cnt);
  gcn_row_kernel<<<NSEG, 128, 0, stream>>>(feats, agg_sum, agg_cnt, Wm, bias, feat_ori, 1);
  // 6) cam_sp_offset + mask indexes
  camoff_kernel<<<NSEG / TB, TB, 0, stream>>>(cam1_sum, seg_cnt, binfo, offv);
  mask_scan_kernel<<<1, 32, 0, stream>>>(offv, mask);
  // 7) augmented branch -> feat_aug
  xaug_gather_kernel<<<(NAUG * Dd) / TB, TB, 0, stream>>>(feats, keep, xaug);
  edge_scatter_kernel<<<(NE2 * Dd) / TB, TB, 0, stream>>>(e2, NE2, xaug, agg2_sum, agg2_cnt);
  gcn_row_kernel<<<NAUG, 128, 0, stream>>>(xaug, agg2_sum, agg2_cnt, Wm, bias, haug, 0);
  pool_scatter_kernel<<<(NAUG * F) / TB, TB, 0, stream>>>(haug, keep, offv, pool_sum, pool_cnt);
  feataug_kernel<<<BS, 128, 0, stream>>>(pool_sum, pool_cnt, feat_aug);
  // 8) q gather and logits
  qgather_kernel<<<(NMASK * F) / TB, TB, 0, stream>>>(feat_ori, mask, qbuf);
  small_logits_kernel<<<(NMASK * BS) / TB, TB, 0, stream>>>(qbuf, feat_aug, out);
  {
    dim3 grid(NMASK / 16, (NQ / 64) / 8);   // 64 x 128 blocks, 8 waves each
    logits_wmma_kernel<<<grid, 256, 0, stream>>>(qbuf, queue, out);
  }
  // 9) cam_sp_reduced + loss scalar
  camred_kernel<<<NMASK / TB, TB, 0, stream>>>(offv, mask, out);
  loss_final_kernel<<<1, 32, 0, stream>>>(acc, out);
}